// LableWiseAttentionLayer_87900800680557
// MI455X (gfx1250) — compile-verified
//
#include <hip/hip_runtime.h>

typedef float v2f __attribute__((ext_vector_type(2)));
typedef float v8f __attribute__((ext_vector_type(8)));

#define TM   128                 // block tile M
#define TN   128                 // block tile N
#define KB   32                  // K slab
#define PAD  4                   // pad in floats -> row stride stays 16B aligned
#define SA   (KB + PAD)          // 36 floats = 144B row stride (A and B-as-[N,K])
#define SBKN (TN + PAD)          // 132 floats = 528B row stride (B-as-[K,N])
#define NT   256                 // 8 waves
#define SMT  256

// ---- CDNA5 async global->LDS copy (ASYNCcnt-tracked), 16B per lane ----
__device__ __forceinline__ void async_copy_b128(unsigned lds_off, const float* gsrc)
{
    asm volatile("global_load_async_to_lds_b128 %0, %1, off"
                 :: "v"(lds_off), "v"(gsrc)
                 : "memory");
}
__device__ __forceinline__ void async_wait_all()
{
    asm volatile("s_wait_asynccnt 0x0" ::: "memory");
}
__device__ __forceinline__ unsigned lds_off_of(const void* p)
{
    // LDS flat aperture keeps the LDS byte offset in the low 32 address bits.
    return (unsigned)(size_t)p;
}

// C[M,N] = op(A[M,K] @ B), optionally fused tanh.
// B_IS_NK = true : B supplied as [N,K] row-major (multiply by Bᵀ; K contiguous)
// B_IS_NK = false: B supplied as [K,N] row-major (N contiguous)
template<bool B_IS_NK, bool APPLY_TANH>
__global__ __launch_bounds__(NT)
void gemm_wmma_f32(const float* __restrict__ A, long lda, long strideA,
                   const float* __restrict__ B, long ldb, long strideB,
                   float* __restrict__ C, long ldc, long strideC,
                   int M, int N, int K)
{
    __shared__ float As[2][TM * SA];
    __shared__ float Bs[2][B_IS_NK ? (TN * SA) : (KB * SBKN)];

    A += (long)blockIdx.z * strideA;
    B += (long)blockIdx.z * strideB;
    C += (long)blockIdx.z * strideC;

    const int tid  = threadIdx.x;
    const int wave = tid >> 5;        // 8 waves (wave32)
    const int lane = tid & 31;
    const int tm   = lane & 15;
    const int th   = lane >> 4;

    const int m0 = blockIdx.y * TM;
    const int n0 = blockIdx.x * TN;
    const int wm = (wave >> 1) * 32;  // wave rows: 4 groups of 32
    const int wn = (wave & 1) * 64;   // wave cols: 2 groups of 64

    // Issue async copies for one K slab into LDS buffer `buf`.
    auto stage = [&](int buf, int k0) {
        #pragma unroll
        for (int r = 0; r < 4; ++r) {          // A tile: 128 rows x 32 (1024 float4)
            int f   = tid + r * NT;
            int row = f >> 3;                  // 8 float4 per row
            int col = (f & 7) << 2;
            int gr  = m0 + row;
            if (gr >= M) gr = M - 1;           // clamp: pollutes only unsaved C rows
            async_copy_b128(lds_off_of(&As[buf][row * SA + col]),
                            A + (long)gr * lda + k0 + col);
        }
        if (B_IS_NK) {
            #pragma unroll
            for (int r = 0; r < 4; ++r) {      // B tile: 128 N-rows x 32 K (K-major in LDS)
                int f = tid + r * NT;
                int n = f >> 3;
                int k = (f & 7) << 2;
                async_copy_b128(lds_off_of(&Bs[buf][n * SA + k]),
                                B + (long)(n0 + n) * ldb + k0 + k);
            }
        } else {
            #pragma unroll
            for (int r = 0; r < 4; ++r) {      // B tile: 32 K-rows x 128 N (N-major in LDS)
                int f = tid + r * NT;
                int k = f >> 5;                // 32 float4 per row
                int n = (f & 31) << 2;
                async_copy_b128(lds_off_of(&Bs[buf][k * SBKN + n]),
                                B + (long)(k0 + k) * ldb + n0 + n);
            }
        }
    };

    v8f acc[2][4] = {};

    int buf = 0;
    stage(buf, 0);
    async_wait_all();
    __syncthreads();

    for (int k0 = 0; k0 < K; k0 += KB) {
        if (k0 + KB < K) stage(buf ^ 1, k0 + KB);   // prefetch next slab (other buffer)

        const float* as = As[buf];
        const float* bs = Bs[buf];
        #pragma unroll
        for (int kk = 0; kk < KB; kk += 4) {
            v2f af[2], bf[4];
            #pragma unroll
            for (int i = 0; i < 2; ++i) {
                // A frag: lane holds row m = lane%16; VGPR j <-> K = 2*(lane/16)+j
                const float* p = &as[(wm + i * 16 + tm) * SA + kk + 2 * th];
                af[i][0] = p[0];
                af[i][1] = p[1];
            }
            #pragma unroll
            for (int j = 0; j < 4; ++j) {
                // B frag: lane holds col n = lane%16; VGPR j' <-> K = 2*(lane/16)+j'
                if (B_IS_NK) {
                    const float* p = &bs[(wn + j * 16 + tm) * SA + kk + 2 * th];
                    bf[j][0] = p[0];
                    bf[j][1] = p[1];
                } else {
                    bf[j][0] = bs[(kk + 2 * th + 0) * SBKN + wn + j * 16 + tm];
                    bf[j][1] = bs[(kk + 2 * th + 1) * SBKN + wn + j * 16 + tm];
                }
            }
            #pragma unroll
            for (int i = 0; i < 2; ++i)
                #pragma unroll
                for (int j = 0; j < 4; ++j)
                    acc[i][j] = __builtin_amdgcn_wmma_f32_16x16x4_f32(
                        false, af[i], false, bf[j], (short)0, acc[i][j], false, false);
        }

        async_wait_all();      // our next-slab copies have landed
        __syncthreads();       // everyone done reading current buffer
        buf ^= 1;
    }

    // Epilogue: C VGPR r <-> row 8*(lane/16)+r, col lane%16
    #pragma unroll
    for (int i = 0; i < 2; ++i) {
        #pragma unroll
        for (int j = 0; j < 4; ++j) {
            #pragma unroll
            for (int r = 0; r < 8; ++r) {
                int row = m0 + wm + i * 16 + 8 * th + r;
                int col = n0 + wn + j * 16 + tm;
                if (row < M) {
                    float v = acc[i][j][r];
                    if (APPLY_TANH) v = tanhf(v);
                    C[(long)row * ldc + col] = v;
                }
            }
        }
    }
}

// In-place softmax over contiguous rows of length 4096 (one block per row).
__global__ __launch_bounds__(SMT)
void softmax_rows4096(float* __restrict__ attn)
{
    __shared__ float red[SMT];
    float* p = attn + (size_t)blockIdx.x * 4096;
    const int tid = threadIdx.x;

    float v[16];
    float mx = -3.402823466e38f;
    #pragma unroll
    for (int i = 0; i < 16; ++i) {
        v[i] = p[tid + i * SMT];
        mx = fmaxf(mx, v[i]);
    }
    red[tid] = mx;
    __syncthreads();
    for (int s = SMT / 2; s > 0; s >>= 1) {
        if (tid < s) red[tid] = fmaxf(red[tid], red[tid + s]);
        __syncthreads();
    }
    mx = red[0];
    __syncthreads();

    float sum = 0.f;
    #pragma unroll
    for (int i = 0; i < 16; ++i) {
        v[i] = __expf(v[i] - mx);
        sum += v[i];
    }
    red[tid] = sum;
    __syncthreads();
    for (int s = SMT / 2; s > 0; s >>= 1) {
        if (tid < s) red[tid] += red[tid + s];
        __syncthreads();
    }
    const float inv = 1.f / red[0];
    #pragma unroll
    for (int i = 0; i < 16; ++i)
        p[tid + i * SMT] = v[i] * inv;
}

extern "C" void kernel_launch(void* const* d_in, const int* in_sizes, int n_in,
                              void* d_out, int out_size, void* d_ws, size_t ws_size,
                              hipStream_t stream)
{
    const int Bn = 4, L = 4096, D = 1024, NL = 8921;

    const float* x  = (const float*)d_in[0];   // [B, L, D]
    const float* W1 = (const float*)d_in[1];   // [D, D]
    const float* W2 = (const float*)d_in[2];   // [NL, D]

    float* out  = (float*)d_out;                     // [B, NL, D]
    float* attn = out + (size_t)Bn * NL * D;         // [B, NL, L]
    float* l1   = (float*)d_ws;                      // [B, L, D] (64 MB)

    dim3 blk(NT);

    // 1) l1 = tanh(x @ W1^T)   M=B*L, N=D, K=D
    gemm_wmma_f32<true, true>
        <<<dim3(D / TN, (Bn * L) / TM, 1), blk, 0, stream>>>(
            x, D, 0, W1, D, 0, l1, D, 0, Bn * L, D, D);

    // 2) attn_raw[b] = W2 @ l1[b]^T -> [NL, L] (scores already transposed)
    gemm_wmma_f32<true, false>
        <<<dim3(L / TN, (NL + TM - 1) / TM, Bn), blk, 0, stream>>>(
            W2, D, 0, l1, D, (long)L * D, attn, L, (long)NL * L, NL, L, D);

    // 3) softmax over L (contiguous rows), in place
    softmax_rows4096<<<dim3(Bn * NL), dim3(SMT), 0, stream>>>(attn);

    // 4) out[b] = attn[b] @ x[b]   M=NL, N=D, K=L
    gemm_wmma_f32<false, false>
        <<<dim3(D / TN, (NL + TM - 1) / TM, Bn), blk, 0, stream>>>(
            attn, L, (long)NL * L, x, D, (long)L * D, out, D, (long)NL * D, NL, D, L);
}